// Codred_84894323573091
// MI455X (gfx1250) — compile-verified
//
#include <hip/hip_runtime.h>

// Problem sizes (fixed by the reference).
#define B_  64
#define S_  512
#define D_  768
#define K_  8
#define R_  277

typedef __attribute__((ext_vector_type(2))) float v2f;
typedef __attribute__((ext_vector_type(8))) float v8f;

// ---------------------------------------------------------------------------
// fp32 WMMA: D(16x16,f32) = A(16x4,f32) * B(4x16,f32) + C  — exact fp32 math.
// ---------------------------------------------------------------------------
__device__ __forceinline__ v8f wmma_f32_16x16x4(v2f a, v2f b, v8f c) {
#if defined(__gfx1250__) && __has_builtin(__builtin_amdgcn_wmma_f32_16x16x4_f32)
  // 8 args: (neg_a, A, neg_b, B, c_mod, C, reuse_a, reuse_b)
  return __builtin_amdgcn_wmma_f32_16x16x4_f32(
      false, a, false, b, (short)0, c, false, false);
#else
  asm volatile("v_wmma_f32_16x16x4_f32 %0, %1, %2, %0"
               : "+v"(c) : "v"(a), "v"(b));
  return c;
#endif
}

// ---------------------------------------------------------------------------
// Kernel 1: single-pass ragged span max-pooling.
// One block per batch row; 768 threads = 1 column each (coalesced along D).
// All 10 span bounds are block-uniform -> range checks become SALU.
// Streams embedding exactly once: 100.7 MB total ~= 4.3 us @ 23.3 TB/s.
// ---------------------------------------------------------------------------
__global__ void span_max_kernel(const float* __restrict__ emb,
                                const int*   __restrict__ h_span,
                                const int*   __restrict__ t_span,
                                const int*   __restrict__ b_spans,
                                float* __restrict__ h_emb,
                                float* __restrict__ t_emb,
                                float* __restrict__ b_emb) {
  const int b = blockIdx.x;
  const int d = threadIdx.x;

  int lo[10], hi[10];
  lo[0] = h_span[b * 2];  hi[0] = h_span[b * 2 + 1];
  lo[1] = t_span[b * 2];  hi[1] = t_span[b * 2 + 1];
#pragma unroll
  for (int k = 0; k < K_; ++k) {
    lo[2 + k] = b_spans[(b * K_ + k) * 2];
    hi[2 + k] = b_spans[(b * K_ + k) * 2 + 1];
  }

  float mx[10];
#pragma unroll
  for (int j = 0; j < 10; ++j) mx[j] = -__builtin_inff();

  const float* e = emb + (size_t)b * S_ * D_ + d;
  for (int s = 0; s < S_; ++s) {
    const float val = e[(size_t)s * D_];
#pragma unroll
    for (int j = 0; j < 10; ++j)
      if (s >= lo[j] && s <= hi[j]) mx[j] = fmaxf(mx[j], val);
  }

  h_emb[b * D_ + d] = mx[0];
  t_emb[b * D_ + d] = mx[1];
#pragma unroll
  for (int k = 0; k < K_; ++k)
    b_emb[(size_t)(b * K_ + k) * D_ + d] = mx[2 + k];
}

// ---------------------------------------------------------------------------
// Kernel 2: C[M,N] = A[M,K] * W[N,K]^T + bias[N], optional ReLU.
// One 16x16 output tile per wave; K stepped by 4 via v_wmma_f32_16x16x4_f32.
// f32 A-frag (16x4) per-lane: lane L holds row (L&15), k = 2*(L>>4)+{0,1}.
// f32 B-frag (4x16) per-lane: lane L holds col (L&15) of W row, same k split.
// => identical addressing for both operands: one float2 load each per step.
// N need not be a multiple of 16: OOB W rows are clamped (results discarded),
// stores are guarded; all branches before the WMMA loop are wave-uniform so
// EXEC is all-1s during every WMMA.
// ---------------------------------------------------------------------------
__global__ void gemm_nt_wmma(const float* __restrict__ A,
                             const float* __restrict__ W,
                             const float* __restrict__ bias,
                             float* __restrict__ C,
                             int M, int N, int Kdim, int relu) {
  const int n_tiles = (N + 15) >> 4;
  const int m_tiles = M >> 4;
  const int wave    = blockIdx.x * (blockDim.x >> 5) + (threadIdx.x >> 5);
  if (wave >= m_tiles * n_tiles) return;  // wave-uniform exit

  const int mt   = wave / n_tiles;
  const int nt   = wave % n_tiles;
  const int lane = threadIdx.x & 31;
  const int half = lane >> 4;   // 0: k 0..1, 1: k 2..3 within each x4 step
  const int r    = lane & 15;

  const float* arow = A + (size_t)(mt * 16 + r) * Kdim + 2 * half;
  int wr = nt * 16 + r;
  if (wr > N - 1) wr = N - 1;   // clamp OOB rows; outputs discarded later
  const float* wrow = W + (size_t)wr * Kdim + 2 * half;

  v8f acc = {};
  for (int k = 0; k < Kdim; k += 16) {
#pragma unroll
    for (int kk = 0; kk < 16; kk += 4) {
      const v2f a  = *(const v2f*)(arow + k + kk);
      const v2f bb = *(const v2f*)(wrow + k + kk);
      acc = wmma_f32_16x16x4(a, bb, acc);
    }
  }

  // C/D layout: lane L, vgpr i -> n = (L&15), m = i + 8*(L>>4).
  const int n = nt * 16 + r;
  if (n < N) {
    const float bv = bias[n];
#pragma unroll
    for (int i = 0; i < 8; ++i) {
      float vout = acc[i] + bv;
      if (relu) vout = fmaxf(vout, 0.0f);
      C[(size_t)(mt * 16 + i + 8 * half) * N + n] = vout;
    }
  }
}

// ---------------------------------------------------------------------------
// Kernel 3: attention over K=8 bridges + fuse pre = u + v + ctx.
// One block per batch row, 256 threads.
// ---------------------------------------------------------------------------
__global__ void attn_kernel(const float* __restrict__ u,
                            const float* __restrict__ v,
                            const float* __restrict__ bi,
                            float* __restrict__ pre) {
  const int b   = blockIdx.x;
  const int tid = threadIdx.x;
  __shared__ float red[256];
  __shared__ float att[K_];

  const float* ub  = u + b * D_;
  const float* vb  = v + b * D_;
  const float* bib = bi + (size_t)b * K_ * D_;

  float dot[K_];
#pragma unroll
  for (int k = 0; k < K_; ++k) {
    float p = 0.f;
    for (int d = tid; d < D_; d += 256) p += ub[d] * bib[k * D_ + d];
    dot[k] = p;
  }

#pragma unroll
  for (int k = 0; k < K_; ++k) {
    red[tid] = dot[k];
    __syncthreads();
    for (int off = 128; off > 0; off >>= 1) {
      if (tid < off) red[tid] += red[tid + off];
      __syncthreads();
    }
    if (tid == 0) att[k] = red[0] * 0.125f;  // 1/sqrt(D_K=64)
    __syncthreads();
  }

  if (tid == 0) {
    float m = att[0];
    for (int k = 1; k < K_; ++k) m = fmaxf(m, att[k]);
    float s = 0.f;
    for (int k = 0; k < K_; ++k) { att[k] = __expf(att[k] - m); s += att[k]; }
    const float inv = 1.0f / s;
    for (int k = 0; k < K_; ++k) att[k] *= inv;
  }
  __syncthreads();

  for (int d = tid; d < D_; d += 256) {
    float c = 0.f;
#pragma unroll
    for (int k = 0; k < K_; ++k) c += att[k] * bib[k * D_ + d];
    pre[b * D_ + d] = ub[d] + vb[d] + c;
  }
}

// ---------------------------------------------------------------------------
extern "C" void kernel_launch(void* const* d_in, const int* in_sizes, int n_in,
                              void* d_out, int out_size, void* d_ws, size_t ws_size,
                              hipStream_t stream) {
  (void)in_sizes; (void)n_in; (void)out_size; (void)ws_size;

  const float* emb     = (const float*)d_in[0];
  const int*   h_span  = (const int*)d_in[1];
  const int*   t_span  = (const int*)d_in[2];
  const int*   b_spans = (const int*)d_in[3];
  const float* wu_w    = (const float*)d_in[4];
  const float* wu_b    = (const float*)d_in[5];
  const float* wv_w    = (const float*)d_in[6];
  const float* wv_b    = (const float*)d_in[7];
  const float* wi_w    = (const float*)d_in[8];
  const float* wi_b    = (const float*)d_in[9];
  const float* ln1_w   = (const float*)d_in[10];
  const float* ln1_b   = (const float*)d_in[11];
  const float* pred_w  = (const float*)d_in[12];
  const float* pred_b  = (const float*)d_in[13];
  float* out = (float*)d_out;

  // Workspace layout (floats); total ~1.03M floats (~4.1 MB).
  float* ws    = (float*)d_ws;
  float* h_emb = ws;                          // [64, 768]
  float* t_emb = h_emb + B_ * D_;             // [64, 768]
  float* b_emb = t_emb + B_ * D_;             // [512, 768]
  float* u     = b_emb + B_ * K_ * D_;        // [64, 768]
  float* vv    = u + B_ * D_;                 // [64, 768]
  float* bi    = vv + B_ * D_;                // [512, 768]
  float* pre   = bi + B_ * K_ * D_;           // [64, 768]
  float* pair  = h_emb;                       // reuse (dead after u-GEMM)

  span_max_kernel<<<B_, D_, 0, stream>>>(emb, h_span, t_span, b_spans,
                                         h_emb, t_emb, b_emb);

  auto launch_gemm = [&](const float* A, const float* W, const float* bias,
                         float* C, int M, int N, int relu) {
    const int waves  = (M / 16) * ((N + 15) / 16);
    const int blocks = (waves + 7) / 8;         // 8 waves (256 threads) / block
    gemm_nt_wmma<<<blocks, 256, 0, stream>>>(A, W, bias, C, M, N, D_, relu);
  };

  launch_gemm(h_emb, wu_w, wu_b, u, B_, D_, 0);          // u  = h @ Wu^T + bu
  launch_gemm(t_emb, wv_w, wv_b, vv, B_, D_, 0);         // v  = t @ Wv^T + bv
  launch_gemm(b_emb, wi_w, wi_b, bi, B_ * K_, D_, 0);    // bi = b @ Wi^T + bi_b

  attn_kernel<<<B_, 256, 0, stream>>>(u, vv, bi, pre);   // pre = u+v+ctx

  launch_gemm(pre, ln1_w, ln1_b, pair, B_, D_, 1);       // pair = relu(...)
  launch_gemm(pair, pred_w, pred_b, out, B_, R_, 0);     // logits [64, 277]
}